// CorticalMicrocircuit_74990128988806
// MI455X (gfx1250) — compile-verified
//
#include <hip/hip_runtime.h>
#include <hip/hip_bf16.h>

// Cortical microcircuit RHS, B = 2^21 independent circuits, y/out layout [12, B].
// Memory-bound: ~201 MB traffic -> ~8.6 us floor at 23.3 TB/s HBM.
// Strategy: 1-thread coefficient-folding kernel (59 params -> 64 SGPR-resident
// fused constants in d_ws), then a wave32 streaming kernel: float4 per thread,
// grid-stride with global_prefetch_b8 software pipelining, v_exp_f32 sigmoids,
// v_rcp_f32 instead of IEEE divide to keep VALU well under the HBM roof.

#define NPAR 59

// ---------------------------------------------------------------------------
// Kernel 1: fold params + drive row into 64 coefficients (runs on 1 thread).
// Layout of coef[64]:
//   [0..5]   2*e0   for pops {P23,P5,St,PV,SST,VIP}
//   [6..11]  v0
//   [12..17] r
//   [18..63] K[46]: dy6(7) dy7(8) dy8(6) dy9(9) dy10(9) dy11(7), signs folded.
// ---------------------------------------------------------------------------
__global__ void cm_coef_kernel(const float* __restrict__ params,
                               const float* __restrict__ lo,
                               const float* __restrict__ hi,
                               const float* __restrict__ t_ptr,
                               const float* __restrict__ drive,
                               int n_t,
                               float* __restrict__ coef) {
    if (threadIdx.x != 0 || blockIdx.x != 0) return;

    float p[NPAR];
    #pragma unroll 1
    for (int i = 0; i < NPAR; ++i)
        p[i] = fminf(fmaxf(params[i], lo[i]), hi[i]);

    const float A_p23_p23 = p[0],  A_p23_p5 = p[1],  A_p23_st = p[2],
                A_p23_PV  = p[3],  A_p23_SST = p[4], A_p23_VIP = p[5],
                a_p23 = p[6], e0_p23 = p[7], v0_p23 = p[8], r_p23 = p[9];
    const float A_p5_p5 = p[10], A_p5_st = p[11], A_p5_PV = p[12], A_p5_SST = p[13],
                a_p5 = p[14], e0_p5 = p[15], v0_p5 = p[16], r_p5 = p[17];
    const float A_st_p23 = p[18], A_st_p5 = p[19], /*A_st_st p[20] unused*/
                A_st_PV = p[21], A_st_SST = p[22],
                a_st = p[23], e0_st = p[24], v0_st = p[25], r_st = p[26];
    const float B_PV_p23 = p[27], B_PV_p5 = p[28], B_PV_st = p[29], B_PV_PV = p[30],
                B_PV_SST = p[31], B_PV_VIP = p[32],
                b_PV = p[33], e0_PV = p[34], v0_PV = p[35], r_PV = p[36];
    const float B_SST_p23 = p[37], B_SST_p5 = p[38], B_SST_PV = p[39],
                B_SST_SST = p[40], B_SST_VIP = p[41],
                b_SST = p[42], e0_SST = p[43], v0_SST = p[44], r_SST = p[45];
    const float B_VIP_PV = p[46], B_VIP_SST = p[47], B_VIP_VIP = p[48],
                b_VIP = p[49], e0_VIP = p[50], v0_VIP = p[51], r_VIP = p[52];
    const float N_p23 = p[53], N_p5 = p[54], N_st = p[55],
                N_PV = p[56], N_SST = p[57], N_VIP = p[58];

    // drive row select: t_idx = clip(t/DT, 0, n_t-1) truncated to int32
    const float t   = t_ptr[0];
    float fidx = fminf(fmaxf(t / 0.002f, 0.0f), (float)(n_t - 1));
    const int ti = (int)fidx;
    const float* d = drive + (size_t)ti * 6;
    const float cc_p23 = d[0], th_p5 = d[1], th_st = d[2],
                th_PV  = d[3], nm_SST = d[4], cc_VIP = d[5];

    // sigmoid constants
    coef[0] = 2.0f * e0_p23; coef[1] = 2.0f * e0_p5; coef[2] = 2.0f * e0_st;
    coef[3] = 2.0f * e0_PV;  coef[4] = 2.0f * e0_SST; coef[5] = 2.0f * e0_VIP;
    coef[6] = v0_p23; coef[7] = v0_p5; coef[8] = v0_st;
    coef[9] = v0_PV;  coef[10] = v0_SST; coef[11] = v0_VIP;
    coef[12] = r_p23; coef[13] = r_p5; coef[14] = r_st;
    coef[15] = r_PV;  coef[16] = r_SST; coef[17] = r_VIP;

    float* K = coef + 18;
    // dy6 (P23 synapse)
    K[0] = cc_p23;
    K[1] = A_st_p23 * a_st * (N_st / N_p23);        // * F_St
    K[2] = A_p23_p23 * a_p23;                       // * F_P23
    K[3] = -(B_PV_p23 * b_PV * (N_PV / N_p23));     // * F_PV
    K[4] = -(B_SST_p23 * b_SST * (N_SST / N_p23));  // * F_SST
    K[5] = -2.0f * a_p23;                           // * dP23
    K[6] = -(a_p23 * a_p23);                        // * P23
    // dy7 (P5)
    K[7]  = th_p5;
    K[8]  = A_st_p5 * a_st * (N_st / N_p5);         // * F_St
    K[9]  = A_p23_p5 * a_p23 * (N_p23 / N_p5);      // * F_P23
    K[10] = A_p5_p5 * a_p5;                         // * F_P5
    K[11] = -(B_PV_p5 * b_PV * (N_PV / N_p5));      // * F_PV
    K[12] = -(B_SST_p5 * b_SST * (N_SST / N_p5));   // * F_SST
    K[13] = -2.0f * a_p5;                           // * dP5
    K[14] = -(a_p5 * a_p5);                         // * P5
    // dy8 (St)
    K[15] = th_st;
    K[16] = A_p23_st * a_p23 * (N_p23 / N_st);      // * F_P23
    K[17] = A_p5_st * a_p5 * (N_p5 / N_st);         // * F_P5
    K[18] = -(B_PV_st * b_PV * (N_PV / N_st));      // * F_PV
    K[19] = -2.0f * a_st;                           // * dSt
    K[20] = -(a_st * a_st);                         // * St
    // dy9 (PV)
    K[21] = th_PV;
    K[22] = A_p23_PV * a_p23 * (N_p23 / N_PV);      // * F_P23
    K[23] = A_p5_PV * a_p5 * (N_p5 / N_PV);         // * F_P5
    K[24] = A_st_PV * a_st * (N_st / N_PV);         // * F_St
    K[25] = -(B_PV_PV * b_PV);                      // * F_PV
    K[26] = -(B_SST_PV * b_SST * (N_SST / N_PV));   // * F_SST
    K[27] = -(B_VIP_PV * b_VIP * (N_VIP / N_PV));   // * F_VIP
    K[28] = -2.0f * b_PV;                           // * dPV
    K[29] = -(b_PV * b_PV);                         // * PV
    // dy10 (SST)
    K[30] = nm_SST;
    K[31] = A_p23_SST * a_p23 * (N_p23 / N_SST);    // * F_P23
    K[32] = A_p5_SST * a_p5 * (N_p5 / N_SST);       // * F_P5
    K[33] = A_st_SST * a_st * (N_st / N_SST);       // * F_St
    K[34] = -(B_PV_SST * b_PV * (N_PV / N_SST));    // * F_PV
    K[35] = -(B_SST_SST * b_SST);                   // * F_SST
    K[36] = -(B_VIP_SST * b_VIP * (N_VIP / N_SST)); // * F_VIP
    K[37] = -2.0f * b_SST;                          // * dSST
    K[38] = -(b_SST * b_SST);                       // * SST
    // dy11 (VIP)
    K[39] = cc_VIP;
    K[40] = A_p23_VIP * a_p23 * (N_p23 / N_VIP);    // * F_P23
    K[41] = -(B_PV_VIP * b_PV * (N_PV / N_VIP));    // * F_PV
    K[42] = -(B_SST_VIP * b_SST * (N_SST / N_VIP)); // * F_SST
    K[43] = -(B_VIP_VIP * b_VIP);                   // * F_VIP
    K[44] = -2.0f * b_VIP;                          // * dVIP
    K[45] = -(b_VIP * b_VIP);                       // * VIP
}

// ---------------------------------------------------------------------------
// Kernel 2: streaming RHS. Each thread owns 4 consecutive circuits (float4
// per row => global_load_b128 / global_store_b128). Grid-stride loop with
// global_prefetch_b8 of the next tile (software pipeline). Coefficients come
// in through a uniform pointer => s_load into SGPRs, broadcast to 32 lanes.
// ---------------------------------------------------------------------------
__global__ __launch_bounds__(256) void cm_rhs_kernel(
    const float* __restrict__ y,
    const float* __restrict__ coef,
    float* __restrict__ out,
    long long Bsz,        // batch size (trailing dim)
    long long nquads) {   // Bsz / 4

    // Uniform coefficient pull (compiler keeps these scalar).
    float te[6], v0[6], rr[6];
    #pragma unroll
    for (int j = 0; j < 6; ++j) {
        te[j] = coef[j];
        v0[j] = coef[6 + j];
        rr[j] = coef[12 + j];
    }
    float K[46];
    #pragma unroll
    for (int j = 0; j < 46; ++j) K[j] = coef[18 + j];

    const long long stride = (long long)gridDim.x * blockDim.x;

    for (long long q = (long long)blockIdx.x * blockDim.x + threadIdx.x;
         q < nquads; q += stride) {
        const long long base = q * 4;

        // Prefetch next grid-stride tile (gfx1250 global_prefetch_b8).
        const long long nbase = base + stride * 4;
        if (nbase < Bsz) {
            #pragma unroll
            for (int r = 0; r < 12; ++r)
                __builtin_prefetch(y + (size_t)r * Bsz + nbase, 0, 3);
        }

        // Load 12 state rows, 128 bits each.
        float4 v[12];
        #pragma unroll
        for (int r = 0; r < 12; ++r)
            v[r] = *reinterpret_cast<const float4*>(y + (size_t)r * Bsz + base);
        const float* vv = reinterpret_cast<const float*>(v);

        float4 o[12];
        float* oo = reinterpret_cast<float*>(o);

        #pragma unroll
        for (int c = 0; c < 4; ++c) {
            float s[12];
            #pragma unroll
            for (int k = 0; k < 12; ++k) s[k] = vv[k * 4 + c];

            // F_X = 2*e0 * rcp(1 + exp(r*(v0 - X)))  -> v_exp_f32 + v_rcp_f32
            float F[6];
            #pragma unroll
            for (int j = 0; j < 6; ++j) {
                const float e = __expf(rr[j] * (v0[j] - s[j]));
                F[j] = te[j] * __builtin_amdgcn_rcpf(1.0f + e);
            }

            const float dy6  = K[0]  + K[1]*F[2] + K[2]*F[0] + K[3]*F[3]
                             + K[4]*F[4] + K[5]*s[6] + K[6]*s[0];
            const float dy7  = K[7]  + K[8]*F[2] + K[9]*F[0] + K[10]*F[1]
                             + K[11]*F[3] + K[12]*F[4] + K[13]*s[7] + K[14]*s[1];
            const float dy8  = K[15] + K[16]*F[0] + K[17]*F[1] + K[18]*F[3]
                             + K[19]*s[8] + K[20]*s[2];
            const float dy9  = K[21] + K[22]*F[0] + K[23]*F[1] + K[24]*F[2]
                             + K[25]*F[3] + K[26]*F[4] + K[27]*F[5]
                             + K[28]*s[9] + K[29]*s[3];
            const float dy10 = K[30] + K[31]*F[0] + K[32]*F[1] + K[33]*F[2]
                             + K[34]*F[3] + K[35]*F[4] + K[36]*F[5]
                             + K[37]*s[10] + K[38]*s[4];
            const float dy11 = K[39] + K[40]*F[0] + K[41]*F[3] + K[42]*F[4]
                             + K[43]*F[5] + K[44]*s[11] + K[45]*s[5];

            // rows 0..5 of output = derivative states (copies of y rows 6..11)
            oo[0*4 + c] = s[6];  oo[1*4 + c] = s[7];  oo[2*4 + c] = s[8];
            oo[3*4 + c] = s[9];  oo[4*4 + c] = s[10]; oo[5*4 + c] = s[11];
            oo[6*4 + c] = dy6;   oo[7*4 + c] = dy7;   oo[8*4 + c] = dy8;
            oo[9*4 + c] = dy9;   oo[10*4 + c] = dy10; oo[11*4 + c] = dy11;
        }

        #pragma unroll
        for (int r = 0; r < 12; ++r)
            *reinterpret_cast<float4*>(out + (size_t)r * Bsz + base) = o[r];
    }
}

extern "C" void kernel_launch(void* const* d_in, const int* in_sizes, int n_in,
                              void* d_out, int out_size, void* d_ws, size_t ws_size,
                              hipStream_t stream) {
    const float* params = (const float*)d_in[0];
    const float* lo     = (const float*)d_in[1];
    const float* hi     = (const float*)d_in[2];
    const float* t      = (const float*)d_in[3];
    const float* y      = (const float*)d_in[4];
    const float* drive  = (const float*)d_in[5];
    float* out  = (float*)d_out;
    float* coef = (float*)d_ws;   // 64 floats of scratch

    const long long Bsz = (long long)in_sizes[4] / 12;
    const int n_t       = in_sizes[5] / 6;
    const long long nquads = Bsz / 4;   // B = 2^21 -> divisible by 4

    cm_coef_kernel<<<1, 32, 0, stream>>>(params, lo, hi, t, drive, n_t, coef);

    const int threads = 256;
    // two grid-stride iterations per thread so the prefetch pipeline has work
    long long want = (nquads + (long long)threads * 2 - 1) / ((long long)threads * 2);
    int blocks = (int)(want < 1 ? 1 : (want > 65535 ? 65535 : want));
    cm_rhs_kernel<<<blocks, threads, 0, stream>>>(y, coef, out, Bsz, nquads);
}